// DAIN_13829794693336
// MI455X (gfx1250) — compile-verified
//
#include <hip/hip_runtime.h>
#include <cstdint>
#include <cstddef>

// ---------------------------------------------------------------------------
// DAIN FilterInterpolation for MI455X (gfx1250).
// Memory-bound op (~456 MB min traffic -> ~20us floor @ 23.3 TB/s).
// gfx1250 paths: global_load_async_to_lds_b32 (+ s_wait_asynccnt) for
// parameter staging, global_prefetch_b8 for streaming planes, NT stores for
// the 217 MB output stream. Interior pixels use b128 row-vector gathers.
// ---------------------------------------------------------------------------

constexpr int Hh   = 288;
constexpr int Ww   = 480;
constexpr int Pp   = Hh * Ww;     // 138240 pixels
constexpr int TILE = 256;         // pixels per block (8 wave32)
constexpr int NBLK = Pp / TILE;   // 540 (exact)
static_assert(Pp % TILE == 0, "tile must divide pixel count");
constexpr int C_CTX = 195;

// LDS staging slots (plane-major, TILE floats per plane)
constexpr int SLOT_OFF0 = 0;   // 2 planes: offset0 x,y
constexpr int SLOT_OFF1 = 2;   // 2 planes: offset1 x,y
constexpr int SLOT_FLT0 = 4;   // 16 planes: filter0
constexpr int SLOT_FLT1 = 20;  // 16 planes: filter1
constexpr int NPLANE    = 36;  // 36 KB LDS per block

typedef float v4f __attribute__((ext_vector_type(4)));

// Dword-aligned 4-float load (may be 16B-unaligned; gfx1250 handles b128 @4B).
__device__ __forceinline__ v4f load4_a4(const float* p) {
  v4f r;
  __builtin_memcpy(&r, p, sizeof(r));
  return r;
}

// Async global->LDS copy, one dword per lane (GVS mode: saddr + 32-bit voffset).
__device__ __forceinline__ void async_to_lds_b32(unsigned lds_addr,
                                                 unsigned byte_off,
                                                 const float* base) {
  asm volatile("global_load_async_to_lds_b32 %0, %1, %2"
               :: "v"(lds_addr), "v"(byte_off), "s"(base)
               : "memory");
}

struct Params {
  float wt[25];    // combined 5x5 weights (filter (x) bilinear)
  int   rowi[5];   // clamped row index * W
  int   coli[5];   // clamped col index
  int   L;         // unclamped left column
  bool  vecok;     // columns L..L+4 all in range -> vector fast path
  float validf;    // 1.0 if source position in range, else 0.0
};

// Compute per-pixel warp parameters from staged LDS data.
// sflow: &stage[slot*TILE + t] (x at [0], y at [TILE])
// sfilt: &stage[slot*TILE + t] (tap k at [k*TILE])
__device__ __forceinline__ void compute_params(const float* sflow,
                                               const float* sfilt,
                                               int x, int y, Params& pr) {
  const float fx = sflow[0];
  const float fy = sflow[TILE];
  const float x2 = (float)x + fx;
  const float y2 = (float)y + fy;
  const bool valid = (x2 >= 0.0f) && (y2 >= 0.0f) &&
                     (x2 <= (float)(Ww - 1)) && (y2 <= (float)(Hh - 1));
  pr.validf = valid ? 1.0f : 0.0f;

  const float ixf = floorf(x2);
  const float iyf = floorf(y2);
  const float a = x2 - ixf;          // sub-pixel x
  const float b = y2 - iyf;          // sub-pixel y
  const int L = (int)ixf - 1;        // left tap
  const int T = (int)iyf - 1;        // top tap
  pr.L = L;
  pr.vecok = (L >= 0) && (L <= Ww - 5);

  const float w00 = (1.0f - a) * (1.0f - b);
  const float w10 = a * (1.0f - b);
  const float w01 = (1.0f - a) * b;
  const float w11 = a * b;

#pragma unroll
  for (int i = 0; i < 5; ++i) {
    int c = L + i;
    c = c < 0 ? 0 : (c > Ww - 1 ? Ww - 1 : c);
    pr.coli[i] = c;
  }
#pragma unroll
  for (int j = 0; j < 5; ++j) {
    int r = T + j;
    r = r < 0 ? 0 : (r > Hh - 1 ? Hh - 1 : r);
    pr.rowi[j] = r * Ww;
  }

  float F[16];
#pragma unroll
  for (int k = 0; k < 16; ++k) F[k] = sfilt[k * TILE];

  // Fold 16-tap filter + 2x2 bilinear into a single 5x5 weight grid.
  // wt[j][i] = w00*F[j][i] + w10*F[j][i-1] + w01*F[j-1][i] + w11*F[j-1][i-1]
#pragma unroll
  for (int j = 0; j < 5; ++j) {
#pragma unroll
    for (int i = 0; i < 5; ++i) {
      float s = 0.0f;
      if (j < 4 && i < 4)   s = fmaf(w00, F[j * 4 + i], s);
      if (j < 4 && i >= 1)  s = fmaf(w10, F[j * 4 + i - 1], s);
      if (j >= 1 && i < 4)  s = fmaf(w01, F[(j - 1) * 4 + i], s);
      if (j >= 1 && i >= 1) s = fmaf(w11, F[(j - 1) * 4 + i - 1], s);
      pr.wt[j * 5 + i] = s;
    }
  }
}

// Scalar 25-tap weighted gather (border pixels; per-column clamped indices).
__device__ __forceinline__ float gather25(const float* __restrict__ plane,
                                          const Params& pr) {
  float a0 = 0.0f, a1 = 0.0f, a2 = 0.0f, a3 = 0.0f, a4 = 0.0f;
#pragma unroll
  for (int j = 0; j < 5; ++j) {
    const float* rp = plane + pr.rowi[j];
    a0 = fmaf(pr.wt[j * 5 + 0], rp[pr.coli[0]], a0);
    a1 = fmaf(pr.wt[j * 5 + 1], rp[pr.coli[1]], a1);
    a2 = fmaf(pr.wt[j * 5 + 2], rp[pr.coli[2]], a2);
    a3 = fmaf(pr.wt[j * 5 + 3], rp[pr.coli[3]], a3);
    a4 = fmaf(pr.wt[j * 5 + 4], rp[pr.coli[4]], a4);
  }
  return ((a0 + a1) + (a2 + a3)) + a4;
}

// Vector fast path: columns contiguous -> 5 x (b128 + b32) row loads.
__device__ __forceinline__ float gather25_vec(const float* __restrict__ plane,
                                              const Params& pr) {
  float acc0 = 0.0f, acc1 = 0.0f;
#pragma unroll
  for (int j = 0; j < 5; ++j) {
    const float* rp = plane + (pr.rowi[j] + pr.L);
    const v4f  v = load4_a4(rp);
    const float e = rp[4];
    acc0 = fmaf(pr.wt[j * 5 + 0], v.x, acc0);
    acc1 = fmaf(pr.wt[j * 5 + 1], v.y, acc1);
    acc0 = fmaf(pr.wt[j * 5 + 2], v.z, acc0);
    acc1 = fmaf(pr.wt[j * 5 + 3], v.w, acc1);
    acc0 = fmaf(pr.wt[j * 5 + 4], e, acc0);
  }
  return acc0 + acc1;
}

__global__ __launch_bounds__(TILE)
void dain_filter_interp_kernel(const float* __restrict__ ref0,
                               const float* __restrict__ ref2,
                               const float* __restrict__ ctx0,
                               const float* __restrict__ ctx2,
                               const float* __restrict__ off0,
                               const float* __restrict__ off1,
                               const float* __restrict__ flt0,
                               const float* __restrict__ flt1,
                               float* __restrict__ out) {
  __shared__ float stage[NPLANE * TILE];  // 36 KB

  const int t  = (int)threadIdx.x;
  const int p  = (int)blockIdx.x * TILE + t;   // pixel index (exact coverage)
  const unsigned voff  = (unsigned)p * 4u;     // per-lane global byte offset
  const unsigned lbase = (unsigned)(uintptr_t)(&stage[0]);  // LDS byte offset

  // ---- Stage all per-pixel parameter planes into LDS via async DMA ----
#pragma unroll
  for (int q = 0; q < 2; ++q)
    async_to_lds_b32(lbase + (unsigned)((SLOT_OFF0 + q) * TILE + t) * 4u,
                     voff, off0 + (size_t)q * Pp);
#pragma unroll
  for (int q = 0; q < 2; ++q)
    async_to_lds_b32(lbase + (unsigned)((SLOT_OFF1 + q) * TILE + t) * 4u,
                     voff, off1 + (size_t)q * Pp);
#pragma unroll
  for (int q = 0; q < 16; ++q)
    async_to_lds_b32(lbase + (unsigned)((SLOT_FLT0 + q) * TILE + t) * 4u,
                     voff, flt0 + (size_t)q * Pp);
#pragma unroll
  for (int q = 0; q < 16; ++q)
    async_to_lds_b32(lbase + (unsigned)((SLOT_FLT1 + q) * TILE + t) * 4u,
                     voff, flt1 + (size_t)q * Pp);

  asm volatile("s_wait_asynccnt 0" ::: "memory");
  __syncthreads();

  const int x = p % Ww;
  const int y = p / Ww;

  Params pr;

  // ---- Flow 0: warp ref0 (hold in regs) and stream ctx0 ----
  compute_params(&stage[SLOT_OFF0 * TILE + t], &stage[SLOT_FLT0 * TILE + t],
                 x, y, pr);

  float r0[3];
#pragma unroll
  for (int c = 0; c < 3; ++c)
    r0[c] = gather25(ref0 + (size_t)c * Pp, pr) * pr.validf;

  if (pr.vecok) {
    for (int c = 0; c < C_CTX; ++c) {
      const float* plane = ctx0 + (size_t)c * Pp;
      if (c + 1 < C_CTX)
        __builtin_prefetch(plane + Pp + pr.rowi[0] + pr.L, 0, 3);
      const float v = gather25_vec(plane, pr) * pr.validf;
      __builtin_nontemporal_store(v, out + (size_t)(3 + c) * Pp + p);
    }
  } else {
    for (int c = 0; c < C_CTX; ++c) {
      const float* plane = ctx0 + (size_t)c * Pp;
      const float v = gather25(plane, pr) * pr.validf;
      __builtin_nontemporal_store(v, out + (size_t)(3 + c) * Pp + p);
    }
  }

  // ---- Flow 1: warp ref2 (blend + store) and stream ctx2 ----
  compute_params(&stage[SLOT_OFF1 * TILE + t], &stage[SLOT_FLT1 * TILE + t],
                 x, y, pr);

#pragma unroll
  for (int c = 0; c < 3; ++c) {
    const float v = gather25(ref2 + (size_t)c * Pp, pr) * pr.validf;
    __builtin_nontemporal_store(0.5f * (r0[c] + v), out + (size_t)c * Pp + p);
  }

  if (pr.vecok) {
    for (int c = 0; c < C_CTX; ++c) {
      const float* plane = ctx2 + (size_t)c * Pp;
      if (c + 1 < C_CTX)
        __builtin_prefetch(plane + Pp + pr.rowi[0] + pr.L, 0, 3);
      const float v = gather25_vec(plane, pr) * pr.validf;
      __builtin_nontemporal_store(v, out + (size_t)(3 + C_CTX + c) * Pp + p);
    }
  } else {
    for (int c = 0; c < C_CTX; ++c) {
      const float* plane = ctx2 + (size_t)c * Pp;
      const float v = gather25(plane, pr) * pr.validf;
      __builtin_nontemporal_store(v, out + (size_t)(3 + C_CTX + c) * Pp + p);
    }
  }
}

extern "C" void kernel_launch(void* const* d_in, const int* in_sizes, int n_in,
                              void* d_out, int out_size, void* d_ws,
                              size_t ws_size, hipStream_t stream) {
  (void)in_sizes; (void)n_in; (void)out_size; (void)d_ws; (void)ws_size;
  const float* ref0 = (const float*)d_in[0];
  const float* ref2 = (const float*)d_in[1];
  const float* ctx0 = (const float*)d_in[2];
  const float* ctx2 = (const float*)d_in[3];
  const float* off0 = (const float*)d_in[4];
  const float* off1 = (const float*)d_in[5];
  const float* flt0 = (const float*)d_in[6];
  const float* flt1 = (const float*)d_in[7];
  float* out = (float*)d_out;

  dain_filter_interp_kernel<<<dim3(NBLK), dim3(TILE), 0, stream>>>(
      ref0, ref2, ctx0, ctx2, off0, off1, flt0, flt1, out);
}